// SCCGNN_61254823576198
// MI455X (gfx1250) — compile-verified
//
#include <hip/hip_runtime.h>
#include <hip/hip_bf16.h>

// Problem constants (from reference): N=4096 rows, K=5 stack, L=3 features.
#define NROW   4096
#define KSTACK 5
#define MK     (NROW * KSTACK)   // 20480 = flattened reduction dim
#define KSPLIT 8                 // reduction split across waves
#define CHUNK  (MK / KSPLIT)     // 2560, divisible by 32

typedef __attribute__((ext_vector_type(2))) float v2f;
typedef __attribute__((ext_vector_type(8))) float v8f;

// ---------------------------------------------------------------------------
// Zero-fill (outputs are accumulated with atomics; harness poisons buffers).
// ---------------------------------------------------------------------------
__global__ void sccgnn_zero(float* __restrict__ p, int n) {
    int i = blockIdx.x * blockDim.x + threadIdx.x;
    if (i < n) p[i] = 0.0f;
}

// ---------------------------------------------------------------------------
// prep: builds the B operands in a WMMA-friendly *interleaved* layout.
//   logical W[m', g] = sum_f act(h[m,f]) * P[f,g,k],  m' = m*K + k
//   stored as  Wb[(m'>>1)*32 + g*2 + (m'&1)]
// so a lane's B fragment (rows K, K+1 of one column) is a contiguous 8-byte
// pair -> single global_load_b64, and the per-lane byte offset advances by a
// constant 256 B per 4-m' step (immediate-offset addressing in the hot loop).
// Columns g >= Lout are zero-padded to 16.
// ---------------------------------------------------------------------------
__global__ void sccgnn_prep(const float* __restrict__ h, int Fin, int relu,
                            const float* __restrict__ P, const float* __restrict__ Q,
                            int Lout, float* __restrict__ Wp, float* __restrict__ Wq) {
    int idx = blockIdx.x * blockDim.x + threadIdx.x;   // 0 .. MK-1  (= m*K + k)
    if (idx >= MK) return;
    int m = idx / KSTACK;
    int k = idx % KSTACK;
    int base = (idx >> 1) * 32 + (idx & 1);            // interleaved row base
    #pragma unroll 1
    for (int g = 0; g < 16; ++g) {
        float sp = 0.0f, sq = 0.0f;
        if (g < Lout) {
            for (int f = 0; f < Fin; ++f) {
                float hv = h[m * Fin + f];
                if (relu) hv = (hv >= 0.0f) ? hv : 0.01f * hv;   // LeakyReLU(0.01)
                sp += hv * P[(f * Lout + g) * KSTACK + k];
                sq += hv * Q[(f * Lout + g) * KSTACK + k];
            }
        }
        Wp[base + g * 2] = sp;
        Wq[base + g * 2] = sq;
    }
}

// ---------------------------------------------------------------------------
// gemm: out[n,g] += sum_{m'} Ld[n,m']*Wp[m',g] + Lu[n,m']*Wq[m',g]
// One wave owns a 16-row tile and a 1/8 slice of the 20480 reduction dim,
// reducing with V_WMMA_F32_16X16X4_F32, then atomically adds the first
// `ncols` columns of its 16x16 fp32 accumulator tile into `out`.
//
// A fragment (f32 16x4): lane l -> M = l%16, VGPR0/1 hold K = 2*(l/16)+{0,1}
//   => one contiguous 8B (v2f) load per lane from the streamed Ld/Lu row.
// B fragment (f32 4x16): lane l -> N = l%16, rows K = 2*(l/16)+{0,1}
//   => one contiguous 8B (v2f) load per lane from interleaved Wb (L2 hit).
// C/D (f32 16x16, v8f):  VGPR v -> M = v + 8*(l/16), N = l%16.
// All hot-loop loads use immediate offsets; pointers bump once per 32 m'.
// ---------------------------------------------------------------------------
__global__ __launch_bounds__(128) void sccgnn_gemm(
    const float* __restrict__ Ld, const float* __restrict__ Lu,
    const float* __restrict__ Wp, const float* __restrict__ Wq,
    float* __restrict__ out, int ncols, int ldout) {
    const int wave  = threadIdx.x >> 5;                 // 0..3
    const int lane  = threadIdx.x & 31;
    const int tile  = blockIdx.x >> 1;                  // 0..255 (16-row tiles)
    const int kchk  = ((blockIdx.x & 1) << 2) | wave;   // 0..7 reduction slice
    const int n0    = tile * 16;
    const int row   = lane & 15;                        // M for A, N for B/C
    const int khalf = (lane >> 4) << 1;                 // 0 or 2

    const int mpBeg = kchk * CHUNK;

    // Per-lane streaming pointers; everything below is +constant offsets.
    const float* ap = Ld + (size_t)(n0 + row) * MK + khalf + mpBeg;
    const float* aq = Lu + (size_t)(n0 + row) * MK + khalf + mpBeg;
    const float* bp = Wp + (size_t)mpBeg * 16 + khalf * 16 + row * 2;
    const float* bq = Wq + (size_t)mpBeg * 16 + khalf * 16 + row * 2;

    v8f c = {};
    for (int it = 0; it < CHUNK / 32; ++it) {           // 80 iterations
        __builtin_prefetch(ap + 1024, 0, 1);            // global_prefetch_b8, 4KB ahead
        __builtin_prefetch(aq + 1024, 0, 1);
        #pragma unroll
        for (int u = 0; u < 32; u += 4) {
            v2f a0 = *(const v2f*)(ap + u);
            v2f a1 = *(const v2f*)(aq + u);
            v2f b0 = *(const v2f*)(bp + u * 16);
            v2f b1 = *(const v2f*)(bq + u * 16);
            c = __builtin_amdgcn_wmma_f32_16x16x4_f32(
                    false, a0, false, b0, (short)0, c, false, false);
            c = __builtin_amdgcn_wmma_f32_16x16x4_f32(
                    false, a1, false, b1, (short)0, c, false, false);
        }
        ap += 32;  aq += 32;
        bp += 32 * 16;  bq += 32 * 16;
    }

    if (row < ncols) {
        const int baseM = n0 + ((lane >> 4) << 3);
        #pragma unroll
        for (int v = 0; v < 8; ++v)
            atomicAdd(&out[(size_t)(baseM + v) * ldout + row], c[v]);
    }
}

// ---------------------------------------------------------------------------
// Launch: zero accumulators, then 3x (prep -> gemm), all on `stream`.
// Workspace: Wp(20480*16) | Wq(20480*16) | h1(4096*3) | h2(4096*3)  ~2.7 MB.
// ---------------------------------------------------------------------------
extern "C" void kernel_launch(void* const* d_in, const int* in_sizes, int n_in,
                              void* d_out, int out_size, void* d_ws, size_t ws_size,
                              hipStream_t stream) {
    const float* Ld = (const float*)d_in[0];
    const float* Lu = (const float*)d_in[1];
    const float* x  = (const float*)d_in[2];
    const float* P1 = (const float*)d_in[3];
    const float* Q1 = (const float*)d_in[4];
    const float* P2 = (const float*)d_in[5];
    const float* Q2 = (const float*)d_in[6];
    const float* P3 = (const float*)d_in[7];
    const float* Q3 = (const float*)d_in[8];
    float* out = (float*)d_out;

    float* Wp = (float*)d_ws;
    float* Wq = Wp + (size_t)MK * 16;
    float* h1 = Wq + (size_t)MK * 16;
    float* h2 = h1 + (size_t)NROW * 3;

    // Zero h1+h2 (contiguous) and the final output.
    {
        int nz = NROW * 3 * 2;
        sccgnn_zero<<<(nz + 255) / 256, 256, 0, stream>>>(h1, nz);
        sccgnn_zero<<<(NROW + 255) / 256, 256, 0, stream>>>(out, NROW);
    }

    const int prepBlocks = (MK + 255) / 256;   // 80
    const int gemmBlocks = 256 * (KSPLIT / 4); // 512 blocks x 4 waves

    // Layer 1: h1 = Ld@(x P1) + Lu@(x Q1)
    sccgnn_prep<<<prepBlocks, 256, 0, stream>>>(x, 1, 0, P1, Q1, 3, Wp, Wq);
    sccgnn_gemm<<<gemmBlocks, 128, 0, stream>>>(Ld, Lu, Wp, Wq, h1, 3, 3);

    // Layer 2: h2 = Ld@(lrelu(h1) P2) + Lu@(lrelu(h1) Q2)
    sccgnn_prep<<<prepBlocks, 256, 0, stream>>>(h1, 3, 1, P2, Q2, 3, Wp, Wq);
    sccgnn_gemm<<<gemmBlocks, 128, 0, stream>>>(Ld, Lu, Wp, Wq, h2, 3, 3);

    // Layer 3: out = Ld@(lrelu(h2) P3) + Lu@(lrelu(h2) Q3)
    sccgnn_prep<<<prepBlocks, 256, 0, stream>>>(h2, 3, 1, P3, Q3, 1, Wp, Wq);
    sccgnn_gemm<<<gemmBlocks, 128, 0, stream>>>(Ld, Lu, Wp, Wq, out, 1, 1);
}